// DistanceNetwork_55095840473351
// MI455X (gfx1250) — compile-verified
//
#include <hip/hip_runtime.h>
#include <cstdint>

// Problem constants (fixed by the reference): S=8192, B=32, D=128, fp32 in/out.
// Roofline: 128MB read (support, once) + 32MB write (out, once) @ 23.3 TB/s
// => ~6.9us floor; ~1 FLOP/byte => purely bandwidth-bound. WMMA deliberately
// unused: the einsum is batched-diagonal, so f32 WMMA (16x16x4 only) wastes
// 16-32x MACs or 3x instructions vs chained v_fmac. The win is data movement:
// async-to-LDS, b128 loads, NT-hinted streaming, coalesced b128 stores.
constexpr int S_DIM   = 8192;
constexpr int B_DIM   = 32;
constexpr int D_DIM   = 128;
constexpr int TS      = 16;    // s-values handled per block (512 rows of 512B)
constexpr int XSTRIDE = 160;   // padded dwords per x-row in LDS: 160 % 64 == 32
                               // => conflict-free ds_load_b128 per 16-lane phase
#define EPS_F 1e-10f

typedef float v4f __attribute__((ext_vector_type(4)));

__global__ __launch_bounds__(256) void distnet_kernel(
    const float* __restrict__ sup,   // [S, B, D]
    const float* __restrict__ xg,    // [B, D]
    float* __restrict__ out)         // [B, S, B]
{
  __shared__ __attribute__((aligned(16))) float x_lds[B_DIM * XSTRIDE]; // 20.0 KB
  __shared__ float tn_lds[B_DIM];                                  // target norms
  __shared__ __attribute__((aligned(16))) float dot_lds[TS * B_DIM]; // dot[sl,b]
  __shared__ float inv_lds[TS * B_DIM];                            // 1/denom[sl,b]

  const int t    = threadIdx.x;
  const int w    = t >> 5;     // wave id 0..7 (wave32)
  const int l    = t & 31;     // lane
  const int ch   = l & 7;      // 16B chunk within a 128B slice
  const int rsel = l >> 3;     // which of 4 rows this lane serves
  const int s0   = blockIdx.x * TS;

  // ---- Early prefetch of this block's 256KB support tile toward GL2 ----
  {
    const float* base = sup + (size_t)s0 * B_DIM * D_DIM;
    #pragma unroll
    for (int k = 0; k < 4; ++k)
      __builtin_prefetch(base + ((size_t)t * 4 + k) * 64, 0, 0); // 256B stride
  }

  // ---- Phase 0a: async-copy input_signal [32x128] into bank-padded LDS ----
  // CDNA5 path: GLOBAL_LOAD_ASYNC_TO_LDS_B128 (tracked by ASYNCcnt), no VGPR
  // round-trip; per-lane LDS addressing lets us insert the bank padding.
  #pragma unroll
  for (int k = 0; k < 4; ++k) {
    int c  = t + k * 256;          // 1024 chunks of 16 bytes
    int b  = c >> 5;               // row (0..31)
    int ci = c & 31;               // chunk-in-row
    unsigned lds_addr = (unsigned)(uintptr_t)&x_lds[b * XSTRIDE + ci * 4];
    const float* gp = xg + (size_t)c * 4;
    asm volatile("global_load_async_to_lds_b128 %0, %1, off"
                 :: "v"(lds_addr), "v"(gp) : "memory");
  }
  asm volatile("s_wait_asynccnt 0" ::: "memory");
  __syncthreads();

  // ---- Phase 0b: wave 0 computes target norms tn[b] = ||x[b,:]|| ----
  if (t < 32) {
    for (int g = 0; g < 8; ++g) {
      int b = g * 4 + rsel;
      float acc = 0.f;
      #pragma unroll
      for (int i = 0; i < 4; ++i) {
        v4f xv = *(const v4f*)&x_lds[b * XSTRIDE + i * 32 + ch * 4];
        acc = fmaf(xv.x, xv.x, acc);
        acc = fmaf(xv.y, xv.y, acc);
        acc = fmaf(xv.z, xv.z, acc);
        acc = fmaf(xv.w, xv.w, acc);
      }
      acc += __shfl_xor(acc, 1);
      acc += __shfl_xor(acc, 2);
      acc += __shfl_xor(acc, 4);
      if (ch == 0) tn_lds[b] = sqrtf(acc);
    }
  }
  __syncthreads();

  // ---- Phase 1: dot[s,b] and inv-denom[s,b] for the 512 rows of this tile.
  // 4 rows per wave-iteration, 8 lanes x float4 per row: each NT b128 load
  // covers four coalesced 128B chunks of consecutive 512B rows (read-once
  // stream -> TH_LOAD_NT keeps it from churning the caches).
  #pragma unroll 2
  for (int k = 0; k < 16; ++k) {
    int g  = w + k * 8;            // row-group 0..127
    int r  = g * 4 + rsel;         // row-in-tile 0..511
    int sl = r >> 5;               // s_local
    int b  = r & 31;
    const float* rowp = sup + (size_t)((s0 + sl) * B_DIM + b) * D_DIM;
    float dacc = 0.f, nacc = 0.f;
    #pragma unroll
    for (int i = 0; i < 4; ++i) {
      v4f a  = __builtin_nontemporal_load((const v4f*)(rowp + i * 32 + ch * 4));
      v4f xv = *(const v4f*)&x_lds[b * XSTRIDE + i * 32 + ch * 4];
      dacc = fmaf(a.x, xv.x, dacc);
      dacc = fmaf(a.y, xv.y, dacc);
      dacc = fmaf(a.z, xv.z, dacc);
      dacc = fmaf(a.w, xv.w, dacc);
      nacc = fmaf(a.x, a.x, nacc);
      nacc = fmaf(a.y, a.y, nacc);
      nacc = fmaf(a.z, a.z, nacc);
      nacc = fmaf(a.w, a.w, nacc);
    }
    // 8-lane butterfly reduction (lanes of one row differ in bits 0..2)
    dacc += __shfl_xor(dacc, 1);
    dacc += __shfl_xor(dacc, 2);
    dacc += __shfl_xor(dacc, 4);
    nacc += __shfl_xor(nacc, 1);
    nacc += __shfl_xor(nacc, 2);
    nacc += __shfl_xor(nacc, 4);
    if (ch == 0) {
      dot_lds[r] = dacc;
      inv_lds[r] = 1.0f / (sqrtf(nacc) * tn_lds[b] + EPS_F);
    }
  }
  __syncthreads();

  // ---- Phase 2: expand out[b, s0+sl, j] = dot[sl, j] * inv[sl, b].
  // Lane (rsel,ch) writes a float4: one b128 wave store = 4 consecutive 128B
  // output rows = 512B contiguous NT stream. dot_lds b128 read is bank-group
  // (8*rsel + ch) mod 16 => conflict-free per 16-lane phase; inv is a 4-way
  // broadcast.
  for (int k = 0; k < 16; ++k) {
    int o  = w * 16 + k;           // 0..127 groups of 4 output rows
    int b  = o >> 2;               // 0..31
    int sg = o & 3;                // group of 4 s-rows
    int sl = sg * 4 + rsel;        // 0..15
    v4f dv = *(const v4f*)&dot_lds[sl * B_DIM + ch * 4];
    float iv = inv_lds[sl * B_DIM + b];
    v4f v = dv * iv;
    __builtin_nontemporal_store(
        v, (v4f*)(out + ((size_t)b * S_DIM + (s0 + sl)) * B_DIM + ch * 4));
  }
}

extern "C" void kernel_launch(void* const* d_in, const int* in_sizes, int n_in,
                              void* d_out, int out_size, void* d_ws, size_t ws_size,
                              hipStream_t stream) {
  (void)in_sizes; (void)n_in; (void)out_size; (void)d_ws; (void)ws_size;
  const float* sup = (const float*)d_in[0];   // support_set  [S,B,D] fp32
  const float* xg  = (const float*)d_in[1];   // input_signal [B,D]  fp32
  float* out = (float*)d_out;                 // [B,S,B] fp32
  distnet_kernel<<<S_DIM / TS, 256, 0, stream>>>(sup, xg, out);
}